// FastPitchTVCGMMLoss_73735998538472
// MI455X (gfx1250) — compile-verified
//
#include <hip/hip_runtime.h>
#include <math.h>
#include <cstdint>

// FastPitch TVCGMM loss for MI455X (gfx1250, wave32).
// Memory-bound streaming reduction: 262MB of mel_out read once -> ~11us floor @23.3TB/s.
// Uses async global->LDS staging (CDNA5 ASYNCcnt path) for coalesced param reads.

#define KMIX 5
#define MIN_VAR 1e-3f
#define HALF_LOG_2PI_3 2.7568155996140185f

#define AS1 __attribute__((address_space(1)))
#define AS3 __attribute__((address_space(3)))

typedef int v2i __attribute__((vector_size(2 * sizeof(int))));   // 64-bit payload type

#if defined(__gfx1250__) && __has_builtin(__builtin_amdgcn_global_load_async_to_lds_b64)
#define HAVE_ASYNC_LDS 1
#else
#define HAVE_ASYNC_LDS 0
#endif

__device__ __forceinline__ float wave_sum32(float v) {
#pragma unroll
  for (int o = 16; o > 0; o >>= 1) v += __shfl_down(v, o, 32);
  return v;
}

__device__ __forceinline__ float softplus_fast(float x) {
  // max(x,0) + log1p(exp(-|x|)) using hw exp/log
  float e = __expf(-fabsf(x));
  return fmaxf(x, 0.0f) + __logf(1.0f + e);
}

// ---------------- main mixture-logprob kernel ----------------
// One thread per (b,t,f) cell; each wave stages its 32 cells' 6400B of params
// through LDS via async copies for coalescing.
__global__ __launch_bounds__(256) void mel_kernel(
    const float* __restrict__ mel_out,
    const unsigned char* __restrict__ dec_mask,
    const float* __restrict__ mel_tgt,
    float* __restrict__ block_part,
    int T, int F, long long total)
{
  const int lane = threadIdx.x & 31;
  const int wid  = threadIdx.x >> 5;
  const long long wave0 = (long long)blockIdx.x * 256 + (wid << 5);
  const long long cell  = wave0 + lane;

#if HAVE_ASYNC_LDS
  __shared__ float tile[8 * 32 * 50];   // 50KB: 8 waves x 32 cells x 50 f32
#endif
  __shared__ float wsum[8];

  const float* cellp = nullptr;
#if HAVE_ASYNC_LDS
  bool staged = (wave0 + 32) <= total;   // wave-uniform
  if (staged) {
    // lane-interleaved float2 copy: 25 async b64 per lane.
    const float* g = mel_out + wave0 * 50 + lane * 2;
    float* l = &tile[wid * 1600 + lane * 2];
#pragma unroll
    for (int i = 0; i < 25; ++i) {
      __builtin_amdgcn_global_load_async_to_lds_b64(
          (AS1 v2i*)(uintptr_t)(g + i * 64),       // global src (flat==global numerically)
          (AS3 v2i*)(v2i*)(l + i * 64),            // LDS dst (addrspacecast)
          0, 0);
    }
#if __has_builtin(__builtin_amdgcn_s_wait_asynccnt)
    __builtin_amdgcn_s_wait_asynccnt(0);
#else
    asm volatile("s_wait_asynccnt 0" ::: "memory");
#endif
    asm volatile("" ::: "memory");   // keep ds reads below the wait
    cellp = &tile[wid * 1600 + lane * 50];
  } else if (cell < total) {
    cellp = mel_out + cell * 50;
  }
#else
  if (cell < total) {
    cellp = mel_out + cell * 50;
    __builtin_prefetch(cellp, 0, 0);   // global_prefetch_b8
  }
#endif

  float local = 0.0f;
  if (cell < total && cellp) {
    const int fi = (int)(cell % F);
    const long long bt = cell / F;
    const int tt = (int)(bt % T);
    const int bb = (int)(bt / T);

    const float* mt = mel_tgt + (long long)bb * F * T;
    float y0 = mt[(long long)fi * T + tt];
    float y1 = (tt + 1 < T) ? mt[(long long)fi * T + tt + 1] : 0.0f;
    float y2 = (fi + 1 < F) ? mt[(long long)(fi + 1) * T + tt] : 0.0f;

    float a[KMIX], w[KMIX];
    float amax = -1e30f, wmax = -1e30f;
#pragma unroll
    for (int k = 0; k < KMIX; ++k) {
      const float* p = cellp + k * 10;
      float wk = p[0];
      float d0 = softplus_fast(p[4]) + MIN_VAR;
      float d1 = softplus_fast(p[5]) + MIN_VAR;
      float d2 = softplus_fast(p[6]) + MIN_VAR;
      float r0 = __builtin_amdgcn_rcpf(d0);
      float r1 = __builtin_amdgcn_rcpf(d1);
      float r2 = __builtin_amdgcn_rcpf(d2);
      float z0 = (y0 - p[1]) * r0;
      float z1 = (y1 - p[2] - p[7] * z0) * r1;
      float z2 = (y2 - p[3] - p[9] * z0 - p[8] * z1) * r2;
      float maha = z0 * z0 + z1 * z1 + z2 * z2;
      float clogp = -0.5f * maha - __logf(d0 * d1 * d2) - HALF_LOG_2PI_3;
      w[k] = wk;
      a[k] = wk + clogp;
      wmax = fmaxf(wmax, wk);
      amax = fmaxf(amax, a[k]);
    }
    float se_a = 0.0f, se_w = 0.0f;
#pragma unroll
    for (int k = 0; k < KMIX; ++k) {
      se_a += __expf(a[k] - amax);
      se_w += __expf(w[k] - wmax);
    }
    float logp = (amax + __logf(se_a)) - (wmax + __logf(se_w));
    float m = dec_mask[(long long)bb * T + tt] ? 1.0f : 0.0f;
    local = logp * m;
  }

  // deterministic block reduction -> per-block partial (no atomics)
  float r = wave_sum32(local);
  if (lane == 0) wsum[wid] = r;
  __syncthreads();
  if (threadIdx.x == 0) {
    float v = 0.0f;
#pragma unroll
    for (int i = 0; i < 8; ++i) v += wsum[i];
    block_part[blockIdx.x] = v;
  }
}

// ---------------- small terms: mask count, duration & pitch MSE ----------------
__global__ __launch_bounds__(256) void aux_kernel(
    const unsigned char* __restrict__ dec_mask, int BT,
    const float* __restrict__ log_dur_pred, const int* __restrict__ dur_tgt,
    const int* __restrict__ dur_lens, const float* __restrict__ pitch_pred,
    const float* __restrict__ pitch_tgt, int N, int BN,
    float* __restrict__ aux_part)
{
  int i = blockIdx.x * 256 + threadIdx.x;
  float msum = (i < BT && dec_mask[i]) ? 1.0f : 0.0f;
  float dsum = 0.0f, psum = 0.0f, dmsum = 0.0f;
  if (i < BN) {
    int b = i / N;
    int n = i - b * N;
    if (n < dur_lens[b]) {
      float ldt = __logf((float)dur_tgt[i] + 1.0f);
      float dd = log_dur_pred[i] - ldt;
      float pp = pitch_tgt[i] - pitch_pred[i];
      dsum = dd * dd;
      psum = pp * pp;
      dmsum = 1.0f;
    }
  }
  __shared__ float tmp[8][4];
  int lane = threadIdx.x & 31, wid = threadIdx.x >> 5;
  msum = wave_sum32(msum);
  dsum = wave_sum32(dsum);
  psum = wave_sum32(psum);
  dmsum = wave_sum32(dmsum);
  if (lane == 0) { tmp[wid][0] = msum; tmp[wid][1] = dsum; tmp[wid][2] = psum; tmp[wid][3] = dmsum; }
  __syncthreads();
  if (threadIdx.x < 4) {
    float v = 0.0f;
#pragma unroll
    for (int w = 0; w < 8; ++w) v += tmp[w][threadIdx.x];
    aux_part[blockIdx.x * 4 + threadIdx.x] = v;
  }
}

// ---------------- deterministic final reductions ----------------
__global__ __launch_bounds__(256) void reduce_mel_kernel(
    const float* __restrict__ part, int n, float* __restrict__ ws)
{
  float s = 0.0f;
  for (int i = threadIdx.x; i < n; i += 256) s += part[i];
  __shared__ float tmp[8];
  int lane = threadIdx.x & 31, wid = threadIdx.x >> 5;
  s = wave_sum32(s);
  if (lane == 0) tmp[wid] = s;
  __syncthreads();
  if (threadIdx.x == 0) {
    float v = 0.0f;
#pragma unroll
    for (int i = 0; i < 8; ++i) v += tmp[i];
    ws[0] = v;   // sum of logp*mask
  }
}

__global__ void finalize_kernel(const float* __restrict__ ws,
                                const float* __restrict__ aux_part, int naux,
                                float* __restrict__ out, float nmel)
{
  float m = 0.0f, d = 0.0f, p = 0.0f, dm = 0.0f;
  for (int i = threadIdx.x; i < naux; i += 32) {
    m  += aux_part[i * 4 + 0];
    d  += aux_part[i * 4 + 1];
    p  += aux_part[i * 4 + 2];
    dm += aux_part[i * 4 + 3];
  }
  m = wave_sum32(m); d = wave_sum32(d); p = wave_sum32(p); dm = wave_sum32(dm);
  if (threadIdx.x == 0) {
    float mel_loss   = -ws[0] / (m * nmel);
    float dur_loss   = d / dm;
    float pitch_loss = p / dm;
    out[0] = mel_loss + pitch_loss * 1.0f + dur_loss * 1.0f;  // PITCH_SCALE=DUR_SCALE=1
    out[1] = mel_loss;
    out[2] = dur_loss;
    out[3] = pitch_loss;
  }
}

extern "C" void kernel_launch(void* const* d_in, const int* in_sizes, int n_in,
                              void* d_out, int out_size, void* d_ws, size_t ws_size,
                              hipStream_t stream) {
  const float* mel_out           = (const float*)d_in[0];
  const unsigned char* dec_mask  = (const unsigned char*)d_in[1];
  // d_in[2] = dur_pred (unused by the loss)
  const float* log_dur_pred      = (const float*)d_in[3];
  const float* pitch_pred        = (const float*)d_in[4];
  const float* mel_tgt           = (const float*)d_in[5];
  const int* dur_tgt             = (const int*)d_in[6];
  const int* dur_lens            = (const int*)d_in[7];
  const float* pitch_tgt         = (const float*)d_in[8];

  const int B  = in_sizes[7];                       // dur_lens: (B,)
  const int N  = in_sizes[2] / B;                   // dur_pred: (B,N)
  const int T  = in_sizes[1] / B;                   // dec_mask: (B,T,1)
  const int F  = (int)((long long)in_sizes[5] / ((long long)B * T)); // mel_tgt: (B,F,T)
  const long long total = (long long)B * T * F;

  float* ws = (float*)d_ws;
  const int nblocks = (int)((total + 255) / 256);
  float* mel_part = ws + 16;
  float* aux_part = mel_part + nblocks;

  const int BT = B * T, BN = B * N;
  const int auxN = BT > BN ? BT : BN;
  const int naux = (auxN + 255) / 256;

  mel_kernel<<<nblocks, 256, 0, stream>>>(mel_out, dec_mask, mel_tgt, mel_part, T, F, total);
  aux_kernel<<<naux, 256, 0, stream>>>(dec_mask, BT, log_dur_pred, dur_tgt, dur_lens,
                                       pitch_pred, pitch_tgt, N, BN, aux_part);
  reduce_mel_kernel<<<1, 256, 0, stream>>>(mel_part, nblocks, ws);
  finalize_kernel<<<1, 32, 0, stream>>>(ws, aux_part, naux, (float*)d_out, (float)F);
}